// GeneratorJournal_46016279609813
// MI455X (gfx1250) — compile-verified
//
#include <hip/hip_runtime.h>
#include <hip/hip_bf16.h>

typedef __attribute__((ext_vector_type(2))) float v2f;
typedef __attribute__((ext_vector_type(8))) float v8f;

#define NQ  6
#define DIM 64
#define SUP 16

// ---------------------------------------------------------------------------
// Prep kernel (1 block): build the 6 quadratic-form matrices B_w from weights
// and store them pre-swizzled into the WMMA B-operand per-lane layout:
//   bsw[kk*64 + lane*2 + {0,1}] = Bflat[(4*kk + 2*(lane>>4) + {0,1})*16 + (lane&15)]
//   where Bflat[16*j + i][w] = B_w[j][i]  (w<6, zero-padded to 16 columns)
// ---------------------------------------------------------------------------
__global__ __launch_bounds__(256) void qsim_prep(const float* __restrict__ w,
                                                 float* __restrict__ bsw) {
  __shared__ float U[SUP][DIM];      // U[j][i] = U_full[i][j<<2] (real circuit)
  __shared__ float Bm[6][SUP][SUP];
  const int tid = threadIdx.x;

  // Phase 1: simulate the fixed circuit on the 16 support basis columns.
  if (tid < SUP) {
    float* st = U[tid];
    for (int i = 0; i < DIM; ++i) st[i] = 0.f;
    st[tid << 2] = 1.f;  // qubits 4,5 = 0 -> full index = j<<2
    for (int layer = 0; layer < 2; ++layer) {
      // CNOT ring: control q, target (q+1)%6 ; in-place disjoint swaps
      for (int q = 0; q < NQ; ++q) {
        const int cm = 1 << (5 - q);
        const int tm = 1 << (5 - ((q + 1) % NQ));
        for (int i = 0; i < DIM; ++i) {
          if ((i & cm) && !(i & tm)) {
            float a = st[i]; st[i] = st[i | tm]; st[i | tm] = a;
          }
        }
      }
      // RY layer: new0 = c*a - s*b ; new1 = s*a + c*b
      for (int q = 0; q < NQ; ++q) {
        const float th = w[layer * NQ + q] * 0.5f;
        const float cc = cosf(th), ss = sinf(th);
        const int msk = 1 << (5 - q);
        for (int i = 0; i < DIM; ++i) {
          if (!(i & msk)) {
            float a = st[i], b = st[i | msk];
            st[i]       = cc * a - ss * b;
            st[i | msk] = ss * a + cc * b;
          }
        }
      }
    }
  }
  __syncthreads();

  // Phase 2: B_w[j][k] = parity(j,k) * u_j*u_k * sum_i z_w(i) U[i,j]U[i,k]
  for (int e = tid; e < 6 * SUP * SUP; e += 256) {
    const int wq = e >> 8;
    const int j = (e >> 4) & 15, k = e & 15;
    const int pj = __popc(j), pk = __popc(k);
    float val = 0.f;
    if (!((pj ^ pk) & 1)) {
      float acc = 0.f;
      const int zm = 1 << (5 - wq);
      for (int i = 0; i < DIM; ++i) {
        const float z = (i & zm) ? -1.f : 1.f;
        acc += z * U[j][i] * U[k][i];
      }
      // u_j from (-i)^popcount: even pc -> (-1)^(pc/2), odd pc -> (-1)^((pc+1)/2)
      const float uj = (pj & 1) ? ((((pj + 1) >> 1) & 1) ? -1.f : 1.f)
                                : (((pj >> 1) & 1) ? -1.f : 1.f);
      const float uk = (pk & 1) ? ((((pk + 1) >> 1) & 1) ? -1.f : 1.f)
                                : (((pk >> 1) & 1) ? -1.f : 1.f);
      val = uj * uk * acc;
    }
    Bm[wq][j][k] = val;
  }
  __syncthreads();

  // Phase 3: emit WMMA B-operand swizzled layout (64 K-chunks x 32 lanes x 2)
  for (int idx = tid; idx < 64 * 32; idx += 256) {
    const int kk = idx >> 5, lane = idx & 31;
    const int h = lane >> 4, n = lane & 15;
    const int j  = kk >> 2;
    const int i0 = ((4 * kk) & 15) + 2 * h;
    float v0 = 0.f, v1 = 0.f;
    if (n < 6) { v0 = Bm[n][j][i0]; v1 = Bm[n][j][i0 + 1]; }
    bsw[idx * 2 + 0] = v0;
    bsw[idx * 2 + 1] = v1;
  }
}

// ---------------------------------------------------------------------------
// Main kernel: per wave, 16-sample tiles. out_tile(16x16) = A(16x256) @ B(256x16)
// via 64 x V_WMMA_F32_16X16X4_F32. A[M][16j+i] = m[M][j]*m[M][i].
// All register indices are compile-time; the only lane-half dependence is the
// qq[2h]/qq[2h+1] pair, resolved with two v_cndmask per tile.
// ---------------------------------------------------------------------------
__global__ __launch_bounds__(256) void qsim_main(const float* __restrict__ x,
                                                 const float* __restrict__ bsw,
                                                 float* __restrict__ out,
                                                 int nTiles, int B) {
  const int lane = threadIdx.x & 31;
  const int h = lane >> 4;          // half selects K-pair within chunk
  const int n = lane & 15;          // A: row M ; B/D: column N
  const int gwave = blockIdx.x * (blockDim.x >> 5) + (threadIdx.x >> 5);
  const int nWaves = gridDim.x * (blockDim.x >> 5);

  // Preload the fixed, pre-swizzled B operand: 64 float2 per lane (16 KB total)
  v2f breg[64];
#pragma unroll
  for (int kk = 0; kk < 64; ++kk)
    breg[kk] = *reinterpret_cast<const v2f*>(bsw + kk * 64 + lane * 2);

  for (int tile = gwave; tile < nTiles; tile += nWaves) {
    const int row = tile * 16 + n;
    const int rowc = row < B ? row : B - 1;  // clamp for partial tail tile
    const float4 xv = *reinterpret_cast<const float4*>(x + (size_t)rowc * 4);

    float c0, s0, c1, s1, c2, s2, c3, s3;
    __sincosf(xv.x * 0.5f, &s0, &c0);
    __sincosf(xv.y * 0.5f, &s1, &c1);
    __sincosf(xv.z * 0.5f, &s2, &c2);
    __sincosf(xv.w * 0.5f, &s3, &c3);

    // Tensor-product factors: m[4t+u] = pp[t]*qq[u]
    float pp[4], qq[4];
    pp[0] = c0 * c1; pp[1] = c0 * s1; pp[2] = s0 * c1; pp[3] = s0 * s1;
    qq[0] = c2 * c3; qq[1] = c2 * s3; qq[2] = s2 * c3; qq[3] = s2 * s3;

    float m[16];
#pragma unroll
    for (int t = 0; t < 4; ++t)
#pragma unroll
      for (int u = 0; u < 4; ++u)
        m[4 * t + u] = pp[t] * qq[u];

    // Per-lane K-pair factors: chunk kk needs m[4t+2h] , m[4t+2h+1]
    const float qa = h ? qq[2] : qq[0];
    const float qb = h ? qq[3] : qq[1];
    v2f ev[4];
#pragma unroll
    for (int t = 0; t < 4; ++t) { ev[t].x = pp[t] * qa; ev[t].y = pp[t] * qb; }

    v8f acc0 = {};
    v8f acc1 = {};
#pragma unroll
    for (int kk = 0; kk < 64; kk += 2) {
      {
        v2f a = ev[kk & 3] * m[kk >> 2];
        acc0 = __builtin_amdgcn_wmma_f32_16x16x4_f32(
            false, a, false, breg[kk], (short)0, acc0, false, false);
      }
      {
        v2f a = ev[(kk + 1) & 3] * m[(kk + 1) >> 2];
        acc1 = __builtin_amdgcn_wmma_f32_16x16x4_f32(
            false, a, false, breg[kk + 1], (short)0, acc1, false, false);
      }
    }

    // D layout: lane (h,n) holds rows M = 8h+v in acc[v], column N = n.
    if (n < 6) {
      const int rbase = tile * 16 + 8 * h;
      if (tile * 16 + 16 <= B) {
#pragma unroll
        for (int v = 0; v < 8; ++v)
          out[(size_t)(rbase + v) * 6 + n] = acc0[v] + acc1[v];
      } else {
#pragma unroll
        for (int v = 0; v < 8; ++v)
          if (rbase + v < B) out[(size_t)(rbase + v) * 6 + n] = acc0[v] + acc1[v];
      }
    }
  }
}

extern "C" void kernel_launch(void* const* d_in, const int* in_sizes, int n_in,
                              void* d_out, int out_size, void* d_ws, size_t ws_size,
                              hipStream_t stream) {
  const float* x = (const float*)d_in[0];     // (B, 4) fp32
  const float* w = (const float*)d_in[1];     // (2, 6) fp32
  float* out = (float*)d_out;                 // (B, 6) fp32
  float* bsw = (float*)d_ws;                  // 16 KB swizzled B operand

  const int B = in_sizes[0] / 4;
  const int tiles = (B + 15) / 16;

  qsim_prep<<<dim3(1), dim3(256), 0, stream>>>(w, bsw);

  // 8 waves/block; aim ~4 tiles per wave to amortize the B-operand preload
  int blocks = (tiles + 31) / 32;
  if (blocks < 1) blocks = 1;
  if (blocks > 2048) blocks = 2048;
  qsim_main<<<dim3(blocks), 256, 0, stream>>>(x, bsw, out, tiles, B);
}